// GGAT_29781303230756
// MI455X (gfx1250) — compile-verified
//
#include <hip/hip_runtime.h>
#include <hip/hip_bf16.h>
#include <math.h>

// GGAT pipeline for MI455X (gfx1250, wave32, WMMA).
// N=8192, NFEAT=512, NHID=128, NCLASS=40, NLAYERS=3.
// - All big GEMMs use V_WMMA_F32_16X16X4_F32 (fp32 in/out == reference numerics;
//   the workload is HBM-bound on the 256MB N x N operands, not FLOP-bound).
// - The N x N score/attention matrices are never materialized: flash-style
//   online softmax fused with the adj@V gate term (adj read exactly once).
// - propagate() streams the adjacency with GLOBAL_LOAD_ASYNC_TO_LDS_B128 into
//   double-buffered LDS tiles (ASYNCcnt pipelined: compute chunk t overlaps the
//   HBM stream of chunk t+1).
// Workspace requirement: ~42 MB (7 * 8192*128 + 8192*384 + 16 floats).

typedef __attribute__((ext_vector_type(2))) float v2f;
typedef __attribute__((ext_vector_type(8))) float v8f;
typedef __attribute__((ext_vector_type(4))) float f4;

#define NV   8192
#define NF   512
#define NH   128
#define NCLS 40
#define NEGV (-9.0e15f)

static __device__ __forceinline__ v8f wmma4(v2f a, v2f b, v8f c) {
  // D = A(16x4,f32) * B(4x16,f32) + C(16x16,f32)
  return __builtin_amdgcn_wmma_f32_16x16x4_f32(
      /*neg_a=*/false, a, /*neg_b=*/false, b,
      /*c_mod=*/(short)0, c, /*reuse_a=*/false, /*reuse_b=*/false);
}

static __device__ __forceinline__ v8f v8f_zero() {
  v8f z = {0.f, 0.f, 0.f, 0.f, 0.f, 0.f, 0.f, 0.f};
  return z;
}

// Async DMA: 16 bytes global -> LDS, tracked by ASYNCcnt (no VGPR round-trip).
// Generic pointers to LDS carry the workgroup-relative offset in the low 32
// bits (aperture in the high bits), which is exactly what VDST wants.
static __device__ __forceinline__ void async_load_b128(const float* g,
                                                       float* lds) {
  unsigned loff = (unsigned)(unsigned long long)lds;
  asm volatile("global_load_async_to_lds_b128 %0, %1, off"
               :
               : "v"(loff), "v"(g)
               : "memory");
}

template <int N>
static __device__ __forceinline__ void wait_asynccnt() {
  asm volatile("s_wait_asynccnt %0" : : "n"(N) : "memory");
}

// ---------------------------------------------------------------------------
// K0: tiny softmaxes for hop_select (3) and the per-layer 2-way gates.
// s[0..2] = softmax(hop_select); s[3..4] = softmax(w[0]); s[5..6] = softmax(w[1])
// ---------------------------------------------------------------------------
__global__ void scalars_kernel(const float* __restrict__ hop,
                               const float* __restrict__ w,
                               float* __restrict__ s) {
  if (threadIdx.x == 0 && blockIdx.x == 0) {
    float m = fmaxf(hop[0], fmaxf(hop[1], hop[2]));
    float e0 = __expf(hop[0] - m), e1 = __expf(hop[1] - m), e2 = __expf(hop[2] - m);
    float d = e0 + e1 + e2;
    s[0] = e0 / d; s[1] = e1 / d; s[2] = e2 / d;
    for (int l = 0; l < 2; ++l) {
      float a = w[l * 2 + 0], b = w[l * 2 + 1];
      float mm = fmaxf(a, b);
      float ea = __expf(a - mm), eb = __expf(b - mm);
      s[3 + l * 2] = ea / (ea + eb);
      s[4 + l * 2] = eb / (ea + eb);
    }
  }
}

// ---------------------------------------------------------------------------
// K1: C[8192 x 128] = A[8192 x K] @ W[128 x K]^T + bias.   (fc1, Qlin, Klin)
// Tile: 128 rows per workgroup (8 waves x 16 rows), full 128 output cols,
// K-chunks of 32 staged in LDS (pitch 36: 16B-aligned, bank-conflict-free).
// ---------------------------------------------------------------------------
__global__ __launch_bounds__(256) void gemm_nt_bias(
    const float* __restrict__ A, const float* __restrict__ W,
    const float* __restrict__ bias, float* __restrict__ C, int K) {
  __shared__ float As[128 * 36];
  __shared__ float Ws[128 * 36];
  const int tid = threadIdx.x;
  const int wave = tid >> 5, lane = tid & 31;
  const int half = lane >> 4, l15 = lane & 15;
  const int rowbase = blockIdx.x * 128;
  const int mw = wave * 16;

  v8f acc[8];
#pragma unroll
  for (int t = 0; t < 8; ++t) acc[t] = v8f_zero();

  for (int k0 = 0; k0 < K; k0 += 32) {
#pragma unroll
    for (int i = 0; i < 4; ++i) {
      int idx = tid + i * 256;
      int r = idx >> 3, c4 = (idx & 7) * 4;
      *(f4*)&As[r * 36 + c4] = *(const f4*)&A[(size_t)(rowbase + r) * K + k0 + c4];
      *(f4*)&Ws[r * 36 + c4] = *(const f4*)&W[(size_t)r * K + k0 + c4];
    }
    __syncthreads();
#pragma unroll
    for (int ks = 0; ks < 8; ++ks) {
      int kk = ks * 4 + half * 2;
      v2f a = *(const v2f*)&As[(mw + l15) * 36 + kk];
#pragma unroll
      for (int t = 0; t < 8; ++t) {
        v2f b = *(const v2f*)&Ws[(t * 16 + l15) * 36 + kk];
        acc[t] = wmma4(a, b, acc[t]);
      }
    }
    __syncthreads();
  }
#pragma unroll
  for (int t = 0; t < 8; ++t) {
    int col = t * 16 + l15;
    float bv = bias[col];
#pragma unroll
    for (int j = 0; j < 8; ++j) {
      int row = rowbase + mw + j + half * 8;
      C[(size_t)row * NH + col] = acc[t][j] + bv;
    }
  }
}

// ---------------------------------------------------------------------------
// K2: Y[8192 x 128] = Adj[8192 x 8192] @ X[8192 x 128]  (NB=1 or 2 B-matrices
// per adjacency pass).  The adjacency (the HBM stream) is fetched with
// GLOBAL_LOAD_ASYNC_TO_LDS_B128 into double-buffered LDS tiles so the DMA of
// chunk t+1 overlaps the WMMA work on chunk t.  X chunks are L2-resident.
// Tile: 64 rows/WG (4 m-waves x 2 n-waves), K-chunks of 32.
// ---------------------------------------------------------------------------
template <int NB>
__global__ __launch_bounds__(256) void propagate(
    const float* __restrict__ Adj,
    const float* __restrict__ X0, const float* __restrict__ X1,
    float* __restrict__ Y0, float* __restrict__ Y1) {
  __shared__ float As[2][64 * 36];
  __shared__ float Xs[NB][32 * NH];
  const int tid = threadIdx.x;
  const int wave = tid >> 5, lane = tid & 31;
  const int half = lane >> 4, l15 = lane & 15;
  const int rowbase = blockIdx.x * 64;
  const int mw = (wave >> 1) * 16;
  const int nw = (wave & 1) * 64;

  v8f acc[NB][4];
#pragma unroll
  for (int b = 0; b < NB; ++b)
#pragma unroll
    for (int t = 0; t < 4; ++t) acc[b][t] = v8f_zero();

  auto loadB = [&](int ksv, v2f (&dst)[NB][4]) {
    int kkv = ksv * 4 + half * 2;
#pragma unroll
    for (int t = 0; t < 4; ++t) {
      int col = nw + t * 16 + l15;
#pragma unroll
      for (int b = 0; b < NB; ++b) {
        dst[b][t].x = Xs[b][kkv * NH + col];
        dst[b][t].y = Xs[b][(kkv + 1) * NH + col];
      }
    }
  };

  // Prologue: async-DMA adjacency chunk 0 into buffer 0 (2 ops per wave).
#pragma unroll
  for (int i = 0; i < 2; ++i) {
    int idx = tid + i * 256;
    int r = idx >> 3, c4 = (idx & 7) * 4;
    async_load_b128(&Adj[(size_t)(rowbase + r) * NV + c4], &As[0][r * 36 + c4]);
  }

  int cur = 0;
  for (int k0 = 0; k0 < NV; k0 += 32) {
    const bool has_next = (k0 + 32 < NV);
    // Kick off the DMA for the NEXT adjacency chunk into the other buffer;
    // the trailing barrier of the previous iteration guarantees it is free.
    if (has_next) {
#pragma unroll
      for (int i = 0; i < 2; ++i) {
        int idx = tid + i * 256;
        int r = idx >> 3, c4 = (idx & 7) * 4;
        async_load_b128(&Adj[(size_t)(rowbase + r) * NV + k0 + 32 + c4],
                        &As[cur ^ 1][r * 36 + c4]);
      }
    }
    // X chunk (L2-resident) -> LDS with regular vector loads; this latency
    // hides under the in-flight adjacency DMA.
#pragma unroll
    for (int b = 0; b < NB; ++b) {
      const float* X = (b == 0) ? X0 : X1;
#pragma unroll
      for (int i = 0; i < 4; ++i) {
        int idx = tid + i * 256;
        int r = idx >> 5, c4 = (idx & 31) * 4;
        *(f4*)&Xs[b][r * NH + c4] = *(const f4*)&X[(size_t)(k0 + r) * NH + c4];
      }
    }
    // Async loads complete in order: waiting to <=2 retires the CURRENT
    // chunk's 2 ops while the next chunk's 2 stay in flight.
    if (has_next) wait_asynccnt<2>(); else wait_asynccnt<0>();
    __syncthreads();

    const float* Ac = As[cur];
    v2f afr[8];
#pragma unroll
    for (int ks = 0; ks < 8; ++ks)
      afr[ks] = *(const v2f*)&Ac[(mw + l15) * 36 + ks * 4 + half * 2];

    // Software-pipeline the B-fragment DS loads one k-step ahead of the WMMAs.
    v2f bcur[NB][4], bnxt[NB][4];
    loadB(0, bcur);
#pragma unroll
    for (int ks = 0; ks < 8; ++ks) {
      if (ks < 7) loadB(ks + 1, bnxt);
#pragma unroll
      for (int t = 0; t < 4; ++t)
#pragma unroll
        for (int b = 0; b < NB; ++b)
          acc[b][t] = wmma4(afr[ks], bcur[b][t], acc[b][t]);
      if (ks < 7) {
#pragma unroll
        for (int t = 0; t < 4; ++t)
#pragma unroll
          for (int b = 0; b < NB; ++b) bcur[b][t] = bnxt[b][t];
      }
    }
    __syncthreads();
    cur ^= 1;
  }
#pragma unroll
  for (int b = 0; b < NB; ++b) {
    float* Y = (b == 0) ? Y0 : Y1;
#pragma unroll
    for (int t = 0; t < 4; ++t)
#pragma unroll
      for (int j = 0; j < 8; ++j)
        Y[(size_t)(rowbase + mw + j + half * 8) * NH + nw + t * 16 + l15] =
            acc[b][t][j];
  }
}

// ---------------------------------------------------------------------------
// K3: flash-style fused masked attention.
// Out = W0 * softmax_row( mask(Q @ K^T, adj>0, NEG) ) @ V + W1 * (adj @ V)
// One workgroup = 16 output rows; 8 waves stripe the 8192 columns with an
// online softmax each; LDS atomic merge at the end. adj is read exactly once.
// ---------------------------------------------------------------------------
__global__ __launch_bounds__(256) void flash_attn(
    const float* __restrict__ Qm, const float* __restrict__ Km,
    const float* __restrict__ V, const float* __restrict__ Adj,
    const float* __restrict__ watt, float* __restrict__ Out) {
  __shared__ float Ps[8][16 * 18];   // per-wave P-tile staging (transpose)
  __shared__ float Aa[8][16 * 18];   // per-wave adj-tile staging
  __shared__ float Oall[16 * NH];
  __shared__ float O2all[16 * NH];
  __shared__ float wmx[8][16];
  __shared__ float gm[16];
  __shared__ float gs[16];

  const int tid = threadIdx.x;
  const int wave = tid >> 5, lane = tid & 31;
  const int half = lane >> 4, l15 = lane & 15;
  const int m0 = blockIdx.x * 16;

  for (int i = tid; i < 16 * NH; i += 256) { Oall[i] = 0.f; O2all[i] = 0.f; }
  if (tid < 16) gs[tid] = 0.f;
  __syncthreads();

  // Q fragments for all 32 k-steps (16 rows x 128) held in registers.
  v2f qf[32];
#pragma unroll
  for (int ks = 0; ks < 32; ++ks) {
    int kk = ks * 4 + half * 2;
    qf[ks] = *(const v2f*)&Qm[(size_t)(m0 + l15) * NH + kk];
  }

  v8f O[8], O2[8];
  float mr[8], sr[8];
#pragma unroll
  for (int t = 0; t < 8; ++t) { O[t] = v8f_zero(); O2[t] = v8f_zero(); }
#pragma unroll
  for (int j = 0; j < 8; ++j) { mr[j] = -__builtin_inff(); sr[j] = 0.f; }

  float* ps = Ps[wave];
  float* as2 = Aa[wave];

  for (int it = 0; it < 64; ++it) {
    const int n0 = (it * 8 + wave) * 16;

    // S = Q @ K^T for this 16x16 tile (k = 128)
    v8f S = v8f_zero();
    const float* kp = &Km[(size_t)(n0 + l15) * NH];
#pragma unroll
    for (int ks = 0; ks < 32; ++ks) {
      int kk = ks * 4 + half * 2;
      v2f bf = *(const v2f*)&kp[kk];
      S = wmma4(qf[ks], bf, S);
    }

    // adjacency tile + mask + online softmax stats
    float av[8], sc[8], P[8];
#pragma unroll
    for (int j = 0; j < 8; ++j) {
      av[j] = Adj[(size_t)(m0 + j + half * 8) * NV + n0 + l15];
      float smv = (av[j] > 0.f) ? S[j] : NEGV;
      float tmax = smv;
      tmax = fmaxf(tmax, __shfl_xor(tmax, 1, 32));
      tmax = fmaxf(tmax, __shfl_xor(tmax, 2, 32));
      tmax = fmaxf(tmax, __shfl_xor(tmax, 4, 32));
      tmax = fmaxf(tmax, __shfl_xor(tmax, 8, 32));
      float mn = fmaxf(mr[j], tmax);
      sc[j] = __expf(mr[j] - mn);
      P[j] = __expf(smv - mn);
      float s = P[j];
      s += __shfl_xor(s, 1, 32);
      s += __shfl_xor(s, 2, 32);
      s += __shfl_xor(s, 4, 32);
      s += __shfl_xor(s, 8, 32);
      sr[j] = sr[j] * sc[j] + s;
      mr[j] = mn;
      ps[(j + half * 8) * 18 + l15] = P[j];
      as2[(j + half * 8) * 18 + l15] = av[j];
    }
#pragma unroll
    for (int t = 0; t < 8; ++t)
#pragma unroll
      for (int j = 0; j < 8; ++j) O[t][j] *= sc[j];

    // O += P @ V_tile ; O2 += adj_tile @ V_tile (same-wave LDS, in-order DS)
#pragma unroll
    for (int ks2 = 0; ks2 < 4; ++ks2) {
      int kk = ks2 * 4 + half * 2;
      v2f ap = *(const v2f*)&ps[l15 * 18 + kk];
      v2f aa = *(const v2f*)&as2[l15 * 18 + kk];
      const float* v0 = &V[(size_t)(n0 + kk) * NH];
      const float* v1 = v0 + NH;
#pragma unroll
      for (int t = 0; t < 8; ++t) {
        int c = t * 16 + l15;
        v2f bf;
        bf.x = v0[c];
        bf.y = v1[c];
        O[t] = wmma4(ap, bf, O[t]);
        O2[t] = wmma4(aa, bf, O2[t]);
      }
    }
  }

  // Merge the 8 per-wave partial softmaxes.
  if (l15 == 0) {
#pragma unroll
    for (int j = 0; j < 8; ++j) wmx[wave][j + half * 8] = mr[j];
  }
  __syncthreads();
  if (tid < 16) {
    float g = wmx[0][tid];
#pragma unroll
    for (int wv = 1; wv < 8; ++wv) g = fmaxf(g, wmx[wv][tid]);
    gm[tid] = g;
  }
  __syncthreads();
#pragma unroll
  for (int j = 0; j < 8; ++j) {
    int r = j + half * 8;
    float f = __expf(mr[j] - gm[r]);
    if (l15 == 0) atomicAdd(&gs[r], sr[j] * f);
#pragma unroll
    for (int t = 0; t < 8; ++t) {
      atomicAdd(&Oall[r * NH + t * 16 + l15], O[t][j] * f);
      atomicAdd(&O2all[r * NH + t * 16 + l15], O2[t][j]);
    }
  }
  __syncthreads();
  const float W0 = watt[0], W1 = watt[1];
  for (int i = tid; i < 16 * NH; i += 256) {
    int r = i >> 7;
    Out[(size_t)(m0 + r) * NH + (i & 127)] =
        W0 * Oall[i] / gs[r] + W1 * O2all[i];
  }
}

// ---------------------------------------------------------------------------
// K4: row-wise L2 normalize * hop weight, scatter into the concat buffer.
// out points at (outs + layer*128); row stride of outs is 384.
// ---------------------------------------------------------------------------
__global__ void norm_rows(const float* __restrict__ in, float* __restrict__ out,
                          const float* __restrict__ hop) {
  __shared__ float r4[4];
  const int row = blockIdx.x, t = threadIdx.x;  // 128 threads
  float v = in[(size_t)row * NH + t];
  float p = v * v;
  p += __shfl_xor(p, 1, 32);
  p += __shfl_xor(p, 2, 32);
  p += __shfl_xor(p, 4, 32);
  p += __shfl_xor(p, 8, 32);
  p += __shfl_xor(p, 16, 32);
  if ((t & 31) == 0) r4[t >> 5] = p;
  __syncthreads();
  float tot = r4[0] + r4[1] + r4[2] + r4[3];
  float d = fmaxf(sqrtf(tot), 1e-12f);
  out[(size_t)row * (3 * NH) + t] = (*hop) * v / d;
}

// ---------------------------------------------------------------------------
// K5: log_softmax( relu(F) @ clf_w^T + clf_b ), 40 classes (tiny; scalar).
// ---------------------------------------------------------------------------
__global__ void classifier(const float* __restrict__ F,
                           const float* __restrict__ Wc,
                           const float* __restrict__ bc,
                           float* __restrict__ outp) {
  __shared__ float Fs[3 * NH];
  __shared__ float L[NCLS];
  __shared__ float red[2];
  const int row = blockIdx.x, t = threadIdx.x;  // 64 threads
#pragma unroll
  for (int i = 0; i < 6; ++i) {
    int idx = t + i * 64;
    Fs[idx] = fmaxf(F[(size_t)row * (3 * NH) + idx], 0.f);
  }
  __syncthreads();
  if (t < NCLS) {
    float a = bc[t];
    const float* w = &Wc[(size_t)t * (3 * NH)];
    for (int k = 0; k < 3 * NH; ++k) a += w[k] * Fs[k];
    L[t] = a;
  }
  __syncthreads();
  if (t == 0) {
    float mx = L[0];
    for (int k = 1; k < NCLS; ++k) mx = fmaxf(mx, L[k]);
    float s = 0.f;
    for (int k = 0; k < NCLS; ++k) s += __expf(L[k] - mx);
    red[0] = mx;
    red[1] = __logf(s);
  }
  __syncthreads();
  if (t < NCLS) outp[(size_t)row * NCLS + t] = L[t] - red[0] - red[1];
}

// ---------------------------------------------------------------------------
extern "C" void kernel_launch(void* const* d_in, const int* in_sizes, int n_in,
                              void* d_out, int out_size, void* d_ws,
                              size_t ws_size, hipStream_t stream) {
  const float* x       = (const float*)d_in[0];
  const float* adj1    = (const float*)d_in[1];
  const float* adj2    = (const float*)d_in[2];
  const float* adj_att = (const float*)d_in[3];
  const float* fc1_w   = (const float*)d_in[4];
  const float* fc1_b   = (const float*)d_in[5];
  const float* Q_w     = (const float*)d_in[6];
  const float* Q_b     = (const float*)d_in[7];
  const float* K_w     = (const float*)d_in[8];
  const float* K_b     = (const float*)d_in[9];
  const float* hop     = (const float*)d_in[10];
  const float* wgate   = (const float*)d_in[11];
  const float* clf_w   = (const float*)d_in[12];
  const float* clf_b   = (const float*)d_in[13];
  float* out = (float*)d_out;

  const size_t NM = (size_t)NV * NH;  // 1M floats
  float* ws    = (float*)d_ws;
  float* scal  = ws;            // 16 floats
  float* tmp   = ws + 16;       // [N, NH]
  float* tatt  = tmp + NM;
  float* qlin  = tatt + NM;
  float* klin  = qlin + NM;
  float* qp    = klin + NM;
  float* kp    = qp + NM;
  float* attn  = kp + NM;
  float* outsb = attn + NM;     // [N, 3*NH]

  scalars_kernel<<<1, 32, 0, stream>>>(hop, wgate, scal);

  // layer 0: plain fc1 -> l2norm -> concat col block 0
  gemm_nt_bias<<<64, 256, 0, stream>>>(x, fc1_w, fc1_b, tmp, NF);
  norm_rows<<<NV, 128, 0, stream>>>(tmp, outsb, scal + 0);

  for (int i = 1; i < 3; ++i) {
    const float* adj = (i == 1) ? adj1 : adj2;
    gemm_nt_bias<<<64, 256, 0, stream>>>(
        x, fc1_w + (size_t)i * NH * NF, fc1_b + i * NH, tmp, NF);
    propagate<1><<<128, 256, 0, stream>>>(adj_att, tmp, nullptr, tatt, nullptr);
    gemm_nt_bias<<<64, 256, 0, stream>>>(
        tatt, Q_w + (size_t)(i - 1) * NH * NH, Q_b + (i - 1) * NH, qlin, NH);
    gemm_nt_bias<<<64, 256, 0, stream>>>(
        tatt, K_w + (size_t)(i - 1) * NH * NH, K_b + (i - 1) * NH, klin, NH);
    propagate<2><<<128, 256, 0, stream>>>(adj_att, qlin, klin, qp, kp);
    flash_attn<<<NV / 16, 256, 0, stream>>>(
        qp, kp, tmp, adj, scal + 3 + (i - 1) * 2, attn);
    norm_rows<<<NV, 128, 0, stream>>>(attn, outsb + i * NH, scal + i);
  }

  classifier<<<NV, 64, 0, stream>>>(outsb, clf_w, clf_b, out);
}